// SobolevLoss_26474178412703
// MI455X (gfx1250) — compile-verified
//
#include <hip/hip_runtime.h>

#define B_ 4
#define T_ 8
#define N_ 16384
#define C_ 4
#define K_ 16
#define EPS_ 1e-8f

typedef __attribute__((ext_vector_type(2))) float v2f;
typedef __attribute__((ext_vector_type(8))) float v8f;

// Exact f32 wave32 sum using V_WMMA_F32_16X16X4_F32 with a ones B-matrix.
// A (16x4 f32): lane m holds A[m,0] (vgpr0) and A[m,1] (vgpr1); lane m+16 holds
// A[m,2], A[m,3]. We put the lane partial in slot 0 and zero in slot 1, so
// D[m,n] = x_m + x_{m+16} for every n. A lane's 8 D vgprs then hold rowsums for
// M=0..7 (lanes 0-15) or M=8..15 (lanes 16-31); adding them and xor-shuffling
// by 16 produces the full 32-lane sum in every lane.
__device__ __forceinline__ float wave_sum32(float x) {
    v2f a; a[0] = x;    a[1] = 0.0f;
    v2f b; b[0] = 1.0f; b[1] = 1.0f;  // all-ones B: layout-independent
    v8f c = {};
    v8f d = __builtin_amdgcn_wmma_f32_16x16x4_f32(
        /*neg_a=*/false, a, /*neg_b=*/false, b,
        /*c_mod=*/(short)0, c, /*reuse_a=*/false, /*reuse_b=*/false);
    float s = d[0] + d[1] + d[2] + d[3] + d[4] + d[5] + d[6] + d[7];
    s += __shfl_xor(s, 16, 32);
    return s;
}

// Reduce two per-thread partials across the block and atomically accumulate
// into two global scalars. Must be called by all threads (uniform flow).
__device__ __forceinline__ void block_accum2(float s0, float s1,
                                             float* g0, float* g1,
                                             float* smem2) {
    if (threadIdx.x == 0) { smem2[0] = 0.0f; smem2[1] = 0.0f; }
    __syncthreads();
    float w0 = wave_sum32(s0);   // full wave, EXEC all-1s
    float w1 = wave_sum32(s1);
    if ((threadIdx.x & 31) == 0) {
        atomicAdd(&smem2[0], w0);   // ds_add_f32
        atomicAdd(&smem2[1], w1);
    }
    __syncthreads();
    if (threadIdx.x == 0) {
        atomicAdd(g0, smem2[0]);    // global_atomic_add_f32
        atomicAdd(g1, smem2[1]);
    }
}

__global__ void k_init(float* __restrict__ ws) {
    if (threadIdx.x < 16) ws[threadIdx.x] = 0.0f;
}

// Partial sums for relative-L2: per batch, sum (pred-target)^2 and target^2.
// grid = (512, B), block = 256; each thread handles one float4 (C=4 aligned).
__global__ void k_rl2_partial(const float4* __restrict__ pred,
                              const float4* __restrict__ targ,
                              float* __restrict__ ws) {
    __shared__ float smem2[2];
    const int b = blockIdx.y;
    const int i = blockIdx.x * blockDim.x + threadIdx.x;       // 0..131071
    const size_t off = (size_t)b * (size_t)(T_ * N_) + (size_t)i;
    float4 p = pred[off];
    float4 t = targ[off];
    float dx = p.x - t.x, dy = p.y - t.y, dz = p.z - t.z, dw = p.w - t.w;
    float sd = dx * dx + dy * dy + dz * dz + dw * dw;
    float st = t.x * t.x + t.y * t.y + t.z * t.z + t.w * t.w;
    block_accum2(sd, st, &ws[b], &ws[4 + b], smem2);
}

// KNN gradient partial sums for BOTH fields in one pass.
// One thread per (b, n): cache 16 indices + 16 reciprocal distances in
// registers, reuse across all T timesteps; C=4 channels move as one float4.
// grid = (N/256, B), block = 256.
__global__ void k_grad_partial(const float4* __restrict__ pred,
                               const float4* __restrict__ targ,
                               const long long* __restrict__ knn,
                               const float* __restrict__ pos,
                               float* __restrict__ ws) {
    __shared__ float smem2[2];
    const int b = blockIdx.y;
    const int n = blockIdx.x * blockDim.x + threadIdx.x;       // 0..N-1

    const long long* kb = knn + ((size_t)b * N_ + (size_t)n) * K_;
    const float*     pb = pos + (size_t)b * N_ * 3;

    int   idx[K_];
    float rinv[K_];
#pragma unroll
    for (int k = 0; k < K_; ++k) idx[k] = (int)kb[k];

    const float px = pb[n * 3 + 0];
    const float py = pb[n * 3 + 1];
    const float pz = pb[n * 3 + 2];
#pragma unroll
    for (int k = 0; k < K_; ++k) {
        const float* pj = pb + (size_t)idx[k] * 3;
        float rx = pj[0] - px, ry = pj[1] - py, rz = pj[2] - pz;
        float d2   = fmaxf(rx * rx + ry * ry + rz * rz, EPS_);
        float dist = fmaxf(sqrtf(d2), EPS_);
        rinv[k] = 1.0f / dist;
    }

    const float4* pf = pred + (size_t)b * (size_t)(T_ * N_);   // float4 units
    const float4* tf = targ + (size_t)b * (size_t)(T_ * N_);

    float sp = 0.0f, st = 0.0f;
    for (int t = 0; t < T_; ++t) {
        const float4* pft = pf + (size_t)t * N_;
        const float4* tft = tf + (size_t)t * N_;
        const float4 pc = pft[n];
        const float4 tc = tft[n];
#pragma unroll
        for (int k = 0; k < K_; ++k) {
            float4 a = pft[idx[k]];
            sp += (fabsf(a.x - pc.x) + fabsf(a.y - pc.y) +
                   fabsf(a.z - pc.z) + fabsf(a.w - pc.w)) * rinv[k];
            float4 u = tft[idx[k]];
            st += (fabsf(u.x - tc.x) + fabsf(u.y - tc.y) +
                   fabsf(u.z - tc.z) + fabsf(u.w - tc.w)) * rinv[k];
        }
    }
    block_accum2(sp, st, &ws[8 + b], &ws[12 + b], smem2);
}

__global__ void k_finalize(const float* __restrict__ ws, float* __restrict__ out) {
    if (threadIdx.x == 0) {
        const float invM = 1.0f / (float)(T_ * (long long)N_ * K_ * C_); // 2^-23
        float rl2 = 0.0f, ge = 0.0f;
#pragma unroll
        for (int b = 0; b < B_; ++b) {
            float num = sqrtf(ws[b]);
            float den = fmaxf(sqrtf(ws[4 + b]), EPS_);
            rl2 += num / den;
            float pg = ws[8 + b] * invM;
            float tg = ws[12 + b] * invM;
            ge += fabsf(pg - tg) / fmaxf(tg, EPS_);
        }
        out[0] = rl2 * (1.0f / B_) + 0.1f * (ge * (1.0f / B_));
    }
}

extern "C" void kernel_launch(void* const* d_in, const int* in_sizes, int n_in,
                              void* d_out, int out_size, void* d_ws, size_t ws_size,
                              hipStream_t stream) {
    const float4*    pred = (const float4*)d_in[0];
    const float4*    targ = (const float4*)d_in[1];
    const long long* knn  = (const long long*)d_in[2];   // int64 per reference
    const float*     pos  = (const float*)d_in[3];
    float* ws  = (float*)d_ws;
    float* out = (float*)d_out;

    k_init<<<1, 32, 0, stream>>>(ws);

    dim3 g1(512, B_);            // 512*256 float4-threads per batch = T*N*C/4
    k_rl2_partial<<<g1, 256, 0, stream>>>(pred, targ, ws);

    dim3 g2(N_ / 256, B_);       // one thread per (b, n)
    k_grad_partial<<<g2, 256, 0, stream>>>(pred, targ, knn, pos, ws);

    k_finalize<<<1, 32, 0, stream>>>(ws, out);
}